// HybridXGModel_14018773254871
// MI455X (gfx1250) — compile-verified
//
#include <hip/hip_runtime.h>
#include <hip/hip_bf16.h>

typedef float v2f __attribute__((ext_vector_type(2)));
typedef float v8f __attribute__((ext_vector_type(8)));

#define HID 64
#define K0PAD 12   // IN_CH=9 padded up to a multiple of 4

// ---------------------------------------------------------------------------
// Degree accumulation: one thread per edge, +1 at dst.
// ---------------------------------------------------------------------------
__global__ void deg_edges_kernel(const int* __restrict__ dst, float* __restrict__ deg, int E) {
    int e = blockIdx.x * blockDim.x + threadIdx.x;
    if (e >= E) return;
    atomicAdd(&deg[dst[e]], 1.0f);
}

// dinv[i] = (deg[i] + 1 /*self loop*/)^-1/2, in place.
__global__ void finish_dinv_kernel(float* __restrict__ deg, int Nn) {
    int i = blockIdx.x * blockDim.x + threadIdx.x;
    if (i >= Nn) return;
    deg[i] = rsqrtf(deg[i] + 1.0f);
}

// ---------------------------------------------------------------------------
// Layer-0 padding: x [N x IC] -> xpad [N x K0PAD] (zero fill), W0 -> [K0PAD x 64]
// ---------------------------------------------------------------------------
__global__ void pad_x_kernel(const float* __restrict__ x, float* __restrict__ xpad,
                             int Nn, int IC) {
    int idx = blockIdx.x * blockDim.x + threadIdx.x;
    if (idx >= Nn * K0PAD) return;
    int r = idx / K0PAD, c = idx - r * K0PAD;
    xpad[idx] = (c < IC) ? x[(size_t)r * IC + c] : 0.0f;
}

__global__ void pad_w0_kernel(const float* __restrict__ W0, float* __restrict__ W0p, int IC) {
    int idx = blockIdx.x * blockDim.x + threadIdx.x;
    if (idx >= K0PAD * HID) return;
    W0p[idx] = (idx < IC * HID) ? W0[idx] : 0.0f;
}

// ---------------------------------------------------------------------------
// fp32 WMMA GEMM: C[M x 64] = A[M x K] * W[K x 64], K compile-time, K % 4 == 0.
// W staged to LDS via CDNA5 async global->LDS DMA, tracked by ASYNCcnt.
// One wave -> one 16x64 tile (4x v_wmma_f32_16x16x4_f32 per K-step, fully
// unrolled). Lane layout per ISA 7.12.2:
//   A frag: lane L holds M=L%16, K = kb + 2*(L>=16) + {0,1}
//   B frag (as loaded): lane L holds K = kb + 2*(L>=16), N = L%16 + {0,1}
//   C/D:    lane L, vgpr j holds M = j + 8*(L>=16), N = L%16
// ---------------------------------------------------------------------------
template <int K>
__global__ void gemm_wmma_f32(const float* __restrict__ A,
                              const float* __restrict__ W,
                              float* __restrict__ C, int M) {
    static_assert(K % 4 == 0, "K must be a multiple of 4");
    __shared__ float Wl[K * HID];

    // --- async stage W (K*64 floats) into LDS: b128 per thread-slice ---
    for (int idx = threadIdx.x; idx < (K * HID) / 4; idx += blockDim.x) {
        unsigned int      lds_addr = (unsigned int)(size_t)(Wl + 4 * idx);
        unsigned long long gaddr   = (unsigned long long)(size_t)(W + 4 * idx);
        asm volatile("global_load_async_to_lds_b128 %0, %1, off"
                     :: "v"(lds_addr), "v"(gaddr) : "memory");
    }
    asm volatile("s_wait_asynccnt 0" ::: "memory");
    __syncthreads();

    const int lane = threadIdx.x & 31;
    const int wave = threadIdx.x >> 5;
    const int row0 = (blockIdx.x * (blockDim.x >> 5) + wave) << 4;
    if (row0 >= M) return;                 // wave-uniform: EXEC stays all-ones
    const int m  = lane & 15;
    const int hi = lane >> 4;              // 0 or 1
    const int kh = hi << 1;                // 0 or 2

    const float* Arow = A + (size_t)(row0 + m) * K;

    v8f acc[4] = {v8f{0}, v8f{0}, v8f{0}, v8f{0}};

#pragma unroll
    for (int kb = 0; kb < K; kb += 4) {
        const int k0 = kb + kh;
        v2f a = *(const v2f*)(Arow + k0);
#pragma unroll
        for (int t = 0; t < 4; ++t) {
            v2f b = *(const v2f*)(Wl + k0 * HID + t * 16 + m);
            acc[t] = __builtin_amdgcn_wmma_f32_16x16x4_f32(
                false, a, false, b, (short)0, acc[t], false, false);
        }
    }

#pragma unroll
    for (int t = 0; t < 4; ++t) {
#pragma unroll
        for (int j = 0; j < 8; ++j) {
            const int r = row0 + j + 8 * hi;
            C[(size_t)r * HID + t * 16 + m] = acc[t][j];
        }
    }
}

// ---------------------------------------------------------------------------
// Edge scatter: 16 lanes per edge, float4 per lane (64 channels).
// acc[dst] += xw[src] * dinv[src]*dinv[dst]
// ---------------------------------------------------------------------------
__global__ void edge_scatter_kernel(const float* __restrict__ xw,
                                    const int* __restrict__ src,
                                    const int* __restrict__ dst,
                                    const float* __restrict__ dinv,
                                    float* __restrict__ acc, int E) {
    int t = blockIdx.x * blockDim.x + threadIdx.x;
    int e = t >> 4;
    if (e >= E) return;
    int c4 = (t & 15) * 4;
    int s = src[e], d = dst[e];
    float nrm = dinv[s] * dinv[d];
    const float4 v = *(const float4*)(xw + (size_t)s * HID + c4);
    float* out = acc + (size_t)d * HID + c4;
    atomicAdd(out + 0, v.x * nrm);
    atomicAdd(out + 1, v.y * nrm);
    atomicAdd(out + 2, v.z * nrm);
    atomicAdd(out + 3, v.w * nrm);
}

// ---------------------------------------------------------------------------
// Epilogue: h = relu(acc + dinv^2 * xw (self loop) + bias)
// ---------------------------------------------------------------------------
__global__ void epilogue_kernel(const float* __restrict__ acc,
                                const float* __restrict__ xw,
                                const float* __restrict__ dinv,
                                const float* __restrict__ bias,
                                float* __restrict__ h, int Nn) {
    int t = blockIdx.x * blockDim.x + threadIdx.x;
    int i = t >> 4;
    if (i >= Nn) return;
    int c4 = (t & 15) * 4;
    float w = dinv[i] * dinv[i];
    const float4 a = *(const float4*)(acc + (size_t)i * HID + c4);
    const float4 x = *(const float4*)(xw  + (size_t)i * HID + c4);
    float4 r;
    r.x = fmaxf(a.x + x.x * w + bias[c4 + 0], 0.0f);
    r.y = fmaxf(a.y + x.y * w + bias[c4 + 1], 0.0f);
    r.z = fmaxf(a.z + x.z * w + bias[c4 + 2], 0.0f);
    r.w = fmaxf(a.w + x.w * w + bias[c4 + 3], 0.0f);
    *(float4*)(h + (size_t)i * HID + c4) = r;
}

// ---------------------------------------------------------------------------
// Global mean pool (accumulate): pooled[batch[i]] += h[i]; cnt[batch[i]] += 1
// ---------------------------------------------------------------------------
__global__ void pool_kernel(const float* __restrict__ h,
                            const int* __restrict__ batch,
                            float* __restrict__ pooled,
                            float* __restrict__ cnt, int Nn) {
    int t = blockIdx.x * blockDim.x + threadIdx.x;
    int i = t >> 4;
    if (i >= Nn) return;
    int c4 = (t & 15) * 4;
    int g = batch[i];
    const float4 v = *(const float4*)(h + (size_t)i * HID + c4);
    float* out = pooled + (size_t)g * HID + c4;
    atomicAdd(out + 0, v.x);
    atomicAdd(out + 1, v.y);
    atomicAdd(out + 2, v.z);
    atomicAdd(out + 3, v.w);
    if ((t & 15) == 0) atomicAdd(&cnt[g], 1.0f);
}

// Z[g, 0:64] = pooled/max(cnt,1);  Z[g, 64:68] = metadata
__global__ void build_z_kernel(const float* __restrict__ pooled,
                               const float* __restrict__ cnt,
                               const float* __restrict__ meta,
                               float* __restrict__ Z, int G, int Kz, int Mm) {
    int t = blockIdx.x * blockDim.x + threadIdx.x;
    if (t >= G * Kz) return;
    int g = t / Kz, c = t - g * Kz;
    float v;
    if (c < HID) v = pooled[(size_t)g * HID + c] / fmaxf(cnt[g], 1.0f);
    else         v = meta[(size_t)g * Mm + (c - HID)];
    Z[t] = v;
}

__global__ void bias_relu_kernel(float* __restrict__ z, const float* __restrict__ b, int total) {
    int t = blockIdx.x * blockDim.x + threadIdx.x;
    if (t >= total) return;
    float v = z[t] + b[t & (HID - 1)];
    z[t] = v > 0.0f ? v : 0.0f;
}

// out[g] = dot(z1[g, 0:64], Wh2) + bh2 ; one wave32 per graph.
__global__ void head2_kernel(const float* __restrict__ z1,
                             const float* __restrict__ Wh2,
                             const float* __restrict__ bh2,
                             float* __restrict__ out) {
    int g = blockIdx.x;
    int lane = threadIdx.x;
    float s = z1[(size_t)g * HID + lane]      * Wh2[lane] +
              z1[(size_t)g * HID + lane + 32] * Wh2[lane + 32];
#pragma unroll
    for (int off = 16; off > 0; off >>= 1) s += __shfl_down(s, off, 32);
    if (lane == 0) out[g] = s + bh2[0];
}

// ---------------------------------------------------------------------------
// Host-side orchestration
// ---------------------------------------------------------------------------
static inline size_t align256(size_t x) { return (x + 255) & ~(size_t)255; }

extern "C" void kernel_launch(void* const* d_in, const int* in_sizes, int n_in,
                              void* d_out, int out_size, void* d_ws, size_t ws_size,
                              hipStream_t stream) {
    const float* x     = (const float*)d_in[0];
    const int*   eidx  = (const int*)  d_in[1];
    const int*   batch = (const int*)  d_in[2];
    const float* meta  = (const float*)d_in[3];
    const float* W0    = (const float*)d_in[4];
    const float* b0    = (const float*)d_in[5];
    const float* W1    = (const float*)d_in[6];
    const float* b1    = (const float*)d_in[7];
    const float* W2    = (const float*)d_in[8];
    const float* b2    = (const float*)d_in[9];
    const float* Wh1   = (const float*)d_in[10];
    const float* bh1   = (const float*)d_in[11];
    const float* Wh2   = (const float*)d_in[12];
    const float* bh2   = (const float*)d_in[13];
    float* out = (float*)d_out;

    const int Nn = in_sizes[2];            // 100000 nodes (batch length)
    const int E  = in_sizes[1] / 2;        // 2,000,000 edges
    const int G  = out_size;               // 1024 graphs (output is [G,1])
    const int Mm = in_sizes[3] / G;        // META = 4
    const int IC = in_sizes[0] / Nn;       // IN_CH = 9
    const int Kz = HID + Mm;               // 68 (must stay multiple of 4)

    const int* src = eidx;                 // edge_index[0]
    const int* dst = eidx + E;             // edge_index[1]

    // --- workspace carve-out ---
    char* ws = (char*)d_ws;
    size_t off = 0;
    float* dinv   = (float*)(ws + off); off += align256((size_t)Nn * 4);
    float* xpad   = (float*)(ws + off); off += align256((size_t)Nn * K0PAD * 4);
    float* W0p    = (float*)(ws + off); off += align256((size_t)K0PAD * HID * 4);
    float* bufA   = (float*)(ws + off); off += align256((size_t)Nn * HID * 4);   // xw
    float* bufB   = (float*)(ws + off); off += align256((size_t)Nn * HID * 4);   // acc
    float* bufC   = (float*)(ws + off); off += align256((size_t)Nn * HID * 4);   // h
    float* pooled = (float*)(ws + off); off += align256((size_t)G * HID * 4);
    float* cnt    = (float*)(ws + off); off += align256((size_t)G * 4);
    float* Z      = (float*)(ws + off); off += align256((size_t)G * Kz * 4);
    float* z1     = (float*)(ws + off); off += align256((size_t)G * HID * 4);
    (void)ws_size;

    const int TB = 256;
    const int edgeBlocks  = (E + TB - 1) / TB;
    const int edge16Blk   = (int)(((size_t)E * 16 + TB - 1) / TB);
    const int node16Blk   = (int)(((size_t)Nn * 16 + TB - 1) / TB);
    const int nodeBlocks  = (Nn + TB - 1) / TB;
    const int gemmBlocksN = (Nn / 16 + 7) / 8;   // 8 waves/block, 16 rows/wave
    const int gemmBlocksG = (G  / 16 + 7) / 8;

    // --- degree / normalization ---
    hipMemsetAsync(dinv, 0, (size_t)Nn * 4, stream);
    deg_edges_kernel<<<edgeBlocks, TB, 0, stream>>>(dst, dinv, E);
    finish_dinv_kernel<<<nodeBlocks, TB, 0, stream>>>(dinv, Nn);

    // --- layer-0 input padding (IN_CH=9 -> 12) ---
    pad_x_kernel<<<(Nn * K0PAD + TB - 1) / TB, TB, 0, stream>>>(x, xpad, Nn, IC);
    pad_w0_kernel<<<(K0PAD * HID + TB - 1) / TB, TB, 0, stream>>>(W0, W0p, IC);

    // --- layer 0 ---
    gemm_wmma_f32<K0PAD><<<gemmBlocksN, TB, 0, stream>>>(xpad, W0p, bufA, Nn);
    hipMemsetAsync(bufB, 0, (size_t)Nn * HID * 4, stream);
    edge_scatter_kernel<<<edge16Blk, TB, 0, stream>>>(bufA, src, dst, dinv, bufB, E);
    epilogue_kernel<<<node16Blk, TB, 0, stream>>>(bufB, bufA, dinv, b0, bufC, Nn);

    // --- layer 1 ---
    gemm_wmma_f32<HID><<<gemmBlocksN, TB, 0, stream>>>(bufC, W1, bufA, Nn);
    hipMemsetAsync(bufB, 0, (size_t)Nn * HID * 4, stream);
    edge_scatter_kernel<<<edge16Blk, TB, 0, stream>>>(bufA, src, dst, dinv, bufB, E);
    epilogue_kernel<<<node16Blk, TB, 0, stream>>>(bufB, bufA, dinv, b1, bufC, Nn);

    // --- layer 2 ---
    gemm_wmma_f32<HID><<<gemmBlocksN, TB, 0, stream>>>(bufC, W2, bufA, Nn);
    hipMemsetAsync(bufB, 0, (size_t)Nn * HID * 4, stream);
    edge_scatter_kernel<<<edge16Blk, TB, 0, stream>>>(bufA, src, dst, dinv, bufB, E);
    epilogue_kernel<<<node16Blk, TB, 0, stream>>>(bufB, bufA, dinv, b2, bufC, Nn);

    // --- global mean pool ---
    hipMemsetAsync(pooled, 0, (size_t)G * HID * 4, stream);
    hipMemsetAsync(cnt,    0, (size_t)G * 4,       stream);
    pool_kernel<<<node16Blk, TB, 0, stream>>>(bufC, batch, pooled, cnt, Nn);

    // --- head MLP ---
    build_z_kernel<<<(G * Kz + TB - 1) / TB, TB, 0, stream>>>(pooled, cnt, meta, Z, G, Kz, Mm);
    gemm_wmma_f32<HID + 4><<<gemmBlocksG, TB, 0, stream>>>(Z, Wh1, z1, G);  // Kz = 68
    bias_relu_kernel<<<(G * HID + TB - 1) / TB, TB, 0, stream>>>(z1, bh1, G * HID);
    head2_kernel<<<G, 32, 0, stream>>>(z1, Wh2, bh2, out);
}